// EABlock_87076166959839
// MI455X (gfx1250) — compile-verified
//
#include <hip/hip_runtime.h>
#include <hip/hip_bf16.h>

typedef __attribute__((ext_vector_type(2))) float v2f;
typedef __attribute__((ext_vector_type(8))) float v8f;

constexpr int C = 256;   // channels
constexpr int D = 64;    // attention dim
constexpr int N = 16384; // h*w
constexpr int B = 16;    // batch

// ---------------------------------------------------------------------------
// Kernel 0: fold linear ops.
//   A[dd][ci]  = sum_o mk_w[dd][o] * conv1_w[o][ci]            (64x256)
//   avec[dd]   = sum_o mk_w[dd][o] * conv1_b[o]                (64)
//   W2[o][dd]  = inv[o] * sum_c conv2_w[o][c] * mv_w[c][dd]    (256x64)
//   b2[o]      = beta[o] - mean[o]*inv[o],  inv = gamma/sqrt(var+eps)
// ---------------------------------------------------------------------------
__global__ __launch_bounds__(256) void k_pre(
    const float* __restrict__ conv1_w, const float* __restrict__ conv1_b,
    const float* __restrict__ mk_w, const float* __restrict__ mv_w,
    const float* __restrict__ conv2_w, const float* __restrict__ gamma,
    const float* __restrict__ beta, const float* __restrict__ mean,
    const float* __restrict__ var,
    float* __restrict__ A, float* __restrict__ avec,
    float* __restrict__ W2, float* __restrict__ b2) {
  int tid = blockIdx.x * 256 + threadIdx.x;
  if (tid < D * C) {                       // A
    int dd = tid >> 8, ci = tid & 255;
    float s = 0.f;
    for (int o = 0; o < C; ++o) s += mk_w[dd * C + o] * conv1_w[o * C + ci];
    A[tid] = s;
  } else if (tid < 2 * D * C) {            // W2 (scaled by BN inv)
    int t = tid - D * C;
    int o = t >> 6, dd = t & 63;
    float s = 0.f;
    for (int c = 0; c < C; ++c) s += conv2_w[o * C + c] * mv_w[c * D + dd];
    float inv = gamma[o] * rsqrtf(var[o] + 1e-5f);
    W2[t] = s * inv;
  } else if (tid < 2 * D * C + D) {        // avec
    int dd = tid - 2 * D * C;
    float s = 0.f;
    for (int o = 0; o < C; ++o) s += mk_w[dd * C + o] * conv1_b[o];
    avec[dd] = s;
  } else if (tid < 2 * D * C + D + C) {    // b2
    int o = tid - 2 * D * C - D;
    float inv = gamma[o] * rsqrtf(var[o] + 1e-5f);
    b2[o] = beta[o] - mean[o] * inv;
  }
}

// ---------------------------------------------------------------------------
// Kernel 1: logits[b][dd][n] = A(64x256) @ x[b](256x16384) + avec
// WMMA f32 16x16x4. WG = 256 threads = 8 waves; each wave owns a 16-column
// strip and all 4 M-tiles (M=64). Grid: (N/128, B).
// Fragment layouts (wave32):
//   A 16x4 : element (m,k) -> vgpr k&1, lane (k>>1)*16 + m
//   B 4x16 : element (k,n) -> vgpr k&1, lane (k>>1)*16 + n
//   C 16x16: vgpr v, lane L -> row v + 8*(L>>4), col L&15
// ---------------------------------------------------------------------------
__global__ __launch_bounds__(256) void k_logits(
    const float* __restrict__ x, const float* __restrict__ A,
    const float* __restrict__ avec, float* __restrict__ logits) {
  const int lane = threadIdx.x & 31;
  const int wave = threadIdx.x >> 5;
  const int b = blockIdx.y;
  const int n0 = (blockIdx.x * 8 + wave) * 16;
  const int half = lane >> 4, lm = lane & 15;

  const float* xb = x + (size_t)b * C * N;
  const float* bp = xb + (size_t)(2 * half) * N + n0 + lm;

  v8f acc[4];
#pragma unroll
  for (int m = 0; m < 4; ++m) acc[m] = (v8f){};

#pragma unroll 4
  for (int k = 0; k < C; k += 4) {
    v2f bf;
    bf.x = bp[(size_t)k * N];
    bf.y = bp[(size_t)k * N + N];
#pragma unroll
    for (int m = 0; m < 4; ++m) {
      v2f af = *(const v2f*)(A + (size_t)(m * 16 + lm) * C + k + 2 * half);
      acc[m] = __builtin_amdgcn_wmma_f32_16x16x4_f32(
          false, af, false, bf, (short)0, acc[m], false, false);
    }
  }

  float* Lb = logits + (size_t)b * D * N;
#pragma unroll
  for (int m = 0; m < 4; ++m) {
#pragma unroll
    for (int v = 0; v < 8; ++v) {
      int row = m * 16 + v + 8 * half;
      Lb[(size_t)row * N + n0 + lm] = acc[m][v] + avec[row];
    }
  }
}

// ---------------------------------------------------------------------------
// Kernel 2a: per-row (b,d) online softmax stats over n=16384.
// One WG of 256 threads per row; single pass (running max + scaled sum).
// ---------------------------------------------------------------------------
__global__ __launch_bounds__(256) void k_rowstats(
    const float* __restrict__ logits, float* __restrict__ rowmax,
    float* __restrict__ rowsuminv) {
  const int row = blockIdx.x;  // b*64 + d, 1024 rows
  const float* p = logits + (size_t)row * N;

  float m = -3.0e38f, s = 0.f;
  for (int i = threadIdx.x; i < N; i += 256) {
    float v = p[i];
    float M = fmaxf(m, v);
    s = s * __expf(m - M) + __expf(v - M);
    m = M;
  }
#pragma unroll
  for (int off = 16; off >= 1; off >>= 1) {
    float m2 = __shfl_xor(m, off, 32);
    float s2 = __shfl_xor(s, off, 32);
    float M = fmaxf(m, m2);
    s = s * __expf(m - M) + s2 * __expf(m2 - M);
    m = M;
  }
  __shared__ float sm[8], ss[8];
  int w = threadIdx.x >> 5, l = threadIdx.x & 31;
  if (l == 0) { sm[w] = m; ss[w] = s; }
  __syncthreads();
  if (threadIdx.x == 0) {
    float M = sm[0], S = ss[0];
    for (int i = 1; i < 8; ++i) {
      float M2 = fmaxf(M, sm[i]);
      S = S * __expf(M - M2) + ss[i] * __expf(sm[i] - M2);
      M = M2;
    }
    rowmax[row] = M;
    rowsuminv[row] = 1.f / S;
  }
}

// ---------------------------------------------------------------------------
// Kernel 2b: attn = softmax(logits) column-normalized over d, IN PLACE.
// One thread per (b,n): 64 strided reads, colsum, 64 strided writes.
// ---------------------------------------------------------------------------
__global__ __launch_bounds__(256) void k_attn(
    float* __restrict__ logits, const float* __restrict__ rowmax,
    const float* __restrict__ rowsuminv) {
  const int b = blockIdx.y;
  const int nn = blockIdx.x * 256 + threadIdx.x;
  float* base = logits + (size_t)b * D * N + nn;
  float vals[D];
  float cs = 0.f;
#pragma unroll
  for (int dd = 0; dd < D; ++dd) {
    float L = base[(size_t)dd * N];
    float v = __expf(L - rowmax[b * D + dd]) * rowsuminv[b * D + dd];
    vals[dd] = v;
    cs += v;
  }
  float inv = 1.f / (1e-9f + cs);
#pragma unroll
  for (int dd = 0; dd < D; ++dd) base[(size_t)dd * N] = vals[dd] * inv;
}

// ---------------------------------------------------------------------------
// Kernel 3: out = relu( W2(256x64) @ attn(64x16384) + b2 + x ), per batch.
// WG = 8 waves, each wave: 16-column strip x 8 M-tiles (half of M=256).
// Grid: (N/128, 2, B).
// ---------------------------------------------------------------------------
__global__ __launch_bounds__(256) void k_out(
    const float* __restrict__ attn, const float* __restrict__ W2,
    const float* __restrict__ b2, const float* __restrict__ x,
    float* __restrict__ out) {
  const int lane = threadIdx.x & 31;
  const int wave = threadIdx.x >> 5;
  const int b = blockIdx.z;
  const int mh = blockIdx.y;  // 0/1 -> rows [128*mh, 128*mh+128)
  const int n0 = (blockIdx.x * 8 + wave) * 16;
  const int half = lane >> 4, lm = lane & 15;

  const float* ab = attn + (size_t)b * D * N + (size_t)(2 * half) * N + n0 + lm;

  v8f acc[8];
#pragma unroll
  for (int m = 0; m < 8; ++m) acc[m] = (v8f){};

#pragma unroll 2
  for (int k = 0; k < D; k += 4) {
    v2f bf;
    bf.x = ab[(size_t)k * N];
    bf.y = ab[(size_t)k * N + N];
#pragma unroll
    for (int m = 0; m < 8; ++m) {
      int row0 = mh * 128 + m * 16;
      v2f af = *(const v2f*)(W2 + (size_t)(row0 + lm) * D + k + 2 * half);
      acc[m] = __builtin_amdgcn_wmma_f32_16x16x4_f32(
          false, af, false, bf, (short)0, acc[m], false, false);
    }
  }

  const float* xb = x + (size_t)b * C * N;
  float* ob = out + (size_t)b * C * N;
#pragma unroll
  for (int m = 0; m < 8; ++m) {
#pragma unroll
    for (int v = 0; v < 8; ++v) {
      int row = mh * 128 + m * 16 + v + 8 * half;
      size_t idx = (size_t)row * N + n0 + lm;
      float val = acc[m][v] + b2[row] + xb[idx];
      ob[idx] = fmaxf(val, 0.f);
    }
  }
}

// ---------------------------------------------------------------------------
// Workspace layout (floats):
//   A:      [0, 16384)            avec:  [16384, 16448)
//   W2:     [16448, 32832)        b2:    [32832, 33088)
//   rowmax: [33088, 34112)        rsinv: [34112, 35136)
//   logits: [35136, 35136+16777216)   (reused in place as attn)
// Total ~67.3 MB.
// ---------------------------------------------------------------------------
extern "C" void kernel_launch(void* const* d_in, const int* in_sizes, int n_in,
                              void* d_out, int out_size, void* d_ws, size_t ws_size,
                              hipStream_t stream) {
  const float* x       = (const float*)d_in[0];
  const float* conv1_w = (const float*)d_in[1];
  const float* conv1_b = (const float*)d_in[2];
  const float* mk_w    = (const float*)d_in[3];
  const float* mv_w    = (const float*)d_in[4];
  const float* conv2_w = (const float*)d_in[5];
  const float* gamma   = (const float*)d_in[6];
  const float* beta    = (const float*)d_in[7];
  const float* mean    = (const float*)d_in[8];
  const float* var     = (const float*)d_in[9];
  float* out = (float*)d_out;

  float* ws = (float*)d_ws;
  float* A      = ws;
  float* avec   = ws + 16384;
  float* W2     = ws + 16448;
  float* b2     = ws + 32832;
  float* rmax   = ws + 33088;
  float* rsinv  = ws + 34112;
  float* logits = ws + 35136;

  k_pre<<<130, 256, 0, stream>>>(conv1_w, conv1_b, mk_w, mv_w, conv2_w,
                                 gamma, beta, mean, var, A, avec, W2, b2);

  dim3 g1(N / 128, B);
  k_logits<<<g1, 256, 0, stream>>>(x, A, avec, logits);

  k_rowstats<<<B * D, 256, 0, stream>>>(logits, rmax, rsinv);

  dim3 g2(N / 256, B);
  k_attn<<<g2, 256, 0, stream>>>(logits, rmax, rsinv);

  dim3 g3(N / 128, 2, B);
  k_out<<<g3, 256, 0, stream>>>(logits, W2, b2, x, out);
}